// KalmanNetLayer_53601191854499
// MI455X (gfx1250) — compile-verified
//
#include <hip/hip_runtime.h>

#define EPSF 1e-6f

typedef __attribute__((ext_vector_type(2))) float v2f;
typedef __attribute__((ext_vector_type(8))) float v8f;
typedef __attribute__((ext_vector_type(4))) int v4i;

#if defined(__AMDGCN__) && __has_builtin(__builtin_amdgcn_global_load_async_to_lds_b128) && __has_builtin(__builtin_amdgcn_s_wait_asynccnt)
#define USE_ASYNC_LDS 1
typedef __attribute__((address_space(1))) v4i* gv4p;   // global int4*
typedef __attribute__((address_space(3))) v4i* lv4p;   // LDS int4*
#else
#define USE_ASYNC_LDS 0
#endif

static __device__ __forceinline__ float wave_sum(float x) {
#pragma unroll
  for (int i = 16; i >= 1; i >>= 1) x += __shfl_xor(x, i, 32);
  return x;
}

// ---------------------------------------------------------------------------
// ae[r, :] = embed[action[r], :]
__global__ void k_gather(const int* __restrict__ action, const float* __restrict__ embed,
                         float* __restrict__ ae, int rows, int cols) {
  int idx = blockIdx.x * blockDim.x + threadIdx.x;
  if (idx >= rows * cols) return;
  int r = idx / cols, c = idx - r * cols;
  ae[idx] = embed[(size_t)action[r] * cols + c];
}

// ---------------------------------------------------------------------------
// C(MxN) = A(MxK) @ B(KxN) (+ bias), row-major fp32, exact-fp32 matrix path
// V_WMMA_F32_16X16X4_F32 (wave32), LDS double-buffered with CDNA5 async copy.
// Block: 256 threads / 8 waves -> 128x64 C tile; wave w owns rows [w*16, w*16+16).
// A staged as [128][20] (stride-20 words: bank-spread, 8B/16B aligned),
// B staged row-major [16][64].
#define KC 16
#define ASTRIDE 20

__global__ void __launch_bounds__(256) k_gemm_wmma(
    const float* __restrict__ A, const float* __restrict__ B,
    const float* __restrict__ bias, float* __restrict__ C,
    int M, int N, int K) {
  __shared__ __align__(16) float As[2][128 * ASTRIDE];
  __shared__ __align__(16) float Bs[2][KC * 64];

  const int nbn = N >> 6;
  const int bm0 = (blockIdx.x / nbn) * 128;
  const int bn0 = (blockIdx.x % nbn) * 64;
  const int t = threadIdx.x;
  const int w = t >> 5;
  const int lane = t & 31;
  const int rr = lane & 15;
  const int kh = (lane >> 4) << 1;  // 0 or 2

  // staging decomposition (per thread): A chunks t and t+256, B chunk t (float4 each)
  const int ar0 = t >> 2, ac0 = (t & 3) << 2;         // A chunk 0: row, col
  const int ar1 = (t + 256) >> 2, ac1 = ac0;          // A chunk 1
  const int br = t >> 4, bc = (t & 15) << 2;          // B chunk

  v8f c0 = {}, c1 = {}, c2 = {}, c3 = {};
  const int NK = K / KC;

  auto issue_stage = [&](int buf, int k0) {
#if USE_ASYNC_LDS
    __builtin_amdgcn_global_load_async_to_lds_b128(
        (gv4p)(A + (size_t)(bm0 + ar0) * K + k0 + ac0),
        (lv4p)&As[buf][ar0 * ASTRIDE + ac0], 0, 0);
    __builtin_amdgcn_global_load_async_to_lds_b128(
        (gv4p)(A + (size_t)(bm0 + ar1) * K + k0 + ac1),
        (lv4p)&As[buf][ar1 * ASTRIDE + ac1], 0, 0);
    __builtin_amdgcn_global_load_async_to_lds_b128(
        (gv4p)(B + (size_t)(k0 + br) * N + bn0 + bc),
        (lv4p)&Bs[buf][br * 64 + bc], 0, 0);
#else
    float4 va0 = *(const float4*)(A + (size_t)(bm0 + ar0) * K + k0 + ac0);
    float4 va1 = *(const float4*)(A + (size_t)(bm0 + ar1) * K + k0 + ac1);
    float4 vb  = *(const float4*)(B + (size_t)(k0 + br) * N + bn0 + bc);
    *(float4*)&As[buf][ar0 * ASTRIDE + ac0] = va0;
    *(float4*)&As[buf][ar1 * ASTRIDE + ac1] = va1;
    *(float4*)&Bs[buf][br * 64 + bc] = vb;
#endif
  };

  issue_stage(0, 0);

  for (int kt = 0; kt < NK; ++kt) {
    const int buf = kt & 1;
#if USE_ASYNC_LDS
    __builtin_amdgcn_s_wait_asynccnt(0);
#endif
    __syncthreads();
    if (kt + 1 < NK) issue_stage(buf ^ 1, (kt + 1) * KC);

    const float* Ab = &As[buf][(w * 16 + rr) * ASTRIDE];
    const float* Bb = &Bs[buf][0];
#pragma unroll
    for (int ks = 0; ks < KC / 4; ++ks) {
      v2f a = *(const v2f*)(Ab + ks * 4 + kh);       // ds_load_b64
      const int bk = ks * 4 + kh;
      v2f b0, b1, b2, b3;
      b0.x = Bb[bk * 64 + rr];      b0.y = Bb[(bk + 1) * 64 + rr];
      b1.x = Bb[bk * 64 + 16 + rr]; b1.y = Bb[(bk + 1) * 64 + 16 + rr];
      b2.x = Bb[bk * 64 + 32 + rr]; b2.y = Bb[(bk + 1) * 64 + 32 + rr];
      b3.x = Bb[bk * 64 + 48 + rr]; b3.y = Bb[(bk + 1) * 64 + 48 + rr];
      c0 = __builtin_amdgcn_wmma_f32_16x16x4_f32(false, a, false, b0, (short)0, c0, false, false);
      c1 = __builtin_amdgcn_wmma_f32_16x16x4_f32(false, a, false, b1, (short)0, c1, false, false);
      c2 = __builtin_amdgcn_wmma_f32_16x16x4_f32(false, a, false, b2, (short)0, c2, false, false);
      c3 = __builtin_amdgcn_wmma_f32_16x16x4_f32(false, a, false, b3, (short)0, c3, false, false);
    }
  }

  // C/D layout: vgpr r -> row = r + 8*(lane>=16), col = lane&15 (per 16x16 tile)
  const int rtop = (lane >> 4) << 3;
  const int col0 = bn0 + rr;
  float bv0 = 0.f, bv1 = 0.f, bv2 = 0.f, bv3 = 0.f;
  if (bias) { bv0 = bias[col0]; bv1 = bias[col0 + 16]; bv2 = bias[col0 + 32]; bv3 = bias[col0 + 48]; }
#pragma unroll
  for (int r = 0; r < 8; ++r) {
    size_t off = (size_t)(bm0 + w * 16 + rtop + r) * N + col0;
    C[off]      = c0[r] + bv0;
    C[off + 16] = c1[r] + bv1;
    C[off + 32] = c2[r] + bv2;
    C[off + 48] = c3[r] + bv3;
  }
}

// ---------------------------------------------------------------------------
// out[r,h] = sigmoid( A[r,:] . W[:,h] ), W is (K x H8) row-major, H8 small (8)
__global__ void k_proj_sigmoid(const float* __restrict__ A, const float* __restrict__ W,
                               float* __restrict__ out, int rows, int K, int H8) {
  int idx = blockIdx.x * blockDim.x + threadIdx.x;
  if (idx >= rows * H8) return;
  int r = idx / H8, h = idx - r * H8;
  const float* a = A + (size_t)r * K;
  float s = 0.f;
  for (int k = 0; k < K; ++k) s += a[k] * W[(size_t)k * H8 + h];
  out[idx] = 1.f / (1.f + expf(-s));
}

// ---------------------------------------------------------------------------
// Per (b,l,h) wave: silu + L2-normalize q and k in place; w = b_v - beta*(b_v.k)*k
__global__ void k_prep(float* __restrict__ q, float* __restrict__ k, float* __restrict__ w,
                       const float* __restrict__ beta, int total) {
  int wave = (blockIdx.x * blockDim.x + threadIdx.x) >> 5;
  int lane = threadIdx.x & 31;
  if (wave >= total) return;
  size_t base = (size_t)wave * 64;
  float q0 = q[base + lane], q1 = q[base + 32 + lane];
  q0 = q0 / (1.f + expf(-q0));
  q1 = q1 / (1.f + expf(-q1));
  float nq = sqrtf(wave_sum(q0 * q0 + q1 * q1)) + EPSF;
  q0 /= nq; q1 /= nq;
  q[base + lane] = q0; q[base + 32 + lane] = q1;

  float k0 = k[base + lane], k1 = k[base + 32 + lane];
  k0 = k0 / (1.f + expf(-k0));
  k1 = k1 / (1.f + expf(-k1));
  float nk = sqrtf(wave_sum(k0 * k0 + k1 * k1)) + EPSF;
  k0 /= nk; k1 /= nk;
  k[base + lane] = k0; k[base + 32 + lane] = k1;

  float bt = beta[wave];
  float bv0 = w[base + lane], bv1 = w[base + 32 + lane];
  float bvk = wave_sum(bv0 * k0 + bv1 * k1);
  w[base + lane]      = bv0 - bt * bvk * k0;
  w[base + 32 + lane] = bv1 - bt * bvk * k1;
}

// ---------------------------------------------------------------------------
// One wave32 per state row (b,h,d): s_new = a1m*(s - beta*(s.k)*k) + bu_d*w + beta*v_d*k
// out[b,l,h,d] = s_new . q    (rowwise rank-1 form of S@decay + update; rows independent)
__global__ void k_scan(const float* __restrict__ q, const float* __restrict__ k,
                       const float* __restrict__ v, const float* __restrict__ bu,
                       const float* __restrict__ w, const float* __restrict__ beta,
                       const float* __restrict__ alpha, const float* __restrict__ mask,
                       const float* __restrict__ carry, float* __restrict__ outp,
                       float* __restrict__ carry_out, int Bn, int Ln, int Hn, int Dn) {
  int wave = (blockIdx.x * blockDim.x + threadIdx.x) >> 5;
  int lane = threadIdx.x & 31;
  int total = Bn * Hn * Dn;
  if (wave >= total) return;
  int d = wave % Dn;
  int h = (wave / Dn) % Hn;
  int b = wave / (Dn * Hn);
  size_t cbase = (size_t)wave * Dn;     // carry layout (B,H,D,D): wave = (b*H+h)*D + d
  float s0 = carry[cbase + lane], s1 = carry[cbase + 32 + lane];
  for (int l = 0; l < Ln; ++l) {
    size_t bl = (size_t)b * Ln + l;
    size_t vbase = bl * (size_t)(Hn * Dn) + (size_t)h * Dn;
    float kk0 = k[vbase + lane], kk1 = k[vbase + 32 + lane];
    float qq0 = q[vbase + lane], qq1 = q[vbase + 32 + lane];
    float ww0 = w[vbase + lane], ww1 = w[vbase + 32 + lane];
    float bt  = beta[bl * Hn + h];
    float a1m = alpha[bl * Hn + h] * (1.f - mask[bl]);
    float vd  = v[vbase + d];
    float bud = bu[vbase + d];
    float t = wave_sum(s0 * kk0 + s1 * kk1);
    float btt = bt * t;
    s0 = a1m * (s0 - btt * kk0) + bud * ww0 + bt * vd * kk0;
    s1 = a1m * (s1 - btt * kk1) + bud * ww1 + bt * vd * kk1;
    float o = wave_sum(s0 * qq0 + s1 * qq1);
    if (lane == 0) outp[vbase + d] = o;
  }
  carry_out[cbase + lane] = s0;
  carry_out[cbase + 32 + lane] = s1;
}

// ---------------------------------------------------------------------------
// RMS-norm over last dim C with learned scale; one wave per row.
__global__ void k_rms(const float* __restrict__ in, const float* __restrict__ scale,
                      float* __restrict__ out, int rows, int C) {
  int wave = (blockIdx.x * blockDim.x + threadIdx.x) >> 5;
  int lane = threadIdx.x & 31;
  if (wave >= rows) return;
  const float* p = in + (size_t)wave * C;
  float ss = 0.f;
  for (int c = lane; c < C; c += 32) { float x = p[c]; ss += x * x; }
  ss = wave_sum(ss);
  float r = rsqrtf(ss / (float)C + EPSF);
  float* o = out + (size_t)wave * C;
  for (int c = lane; c < C; c += 32) o[c] = p[c] * r * scale[c];
}

// ---------------------------------------------------------------------------
extern "C" void kernel_launch(void* const* d_in, const int* in_sizes, int n_in,
                              void* d_out, int out_size, void* d_ws, size_t ws_size,
                              hipStream_t stream) {
  (void)in_sizes; (void)n_in; (void)out_size; (void)ws_size;
  const float* x       = (const float*)d_in[0];
  const int*   action  = (const int*)  d_in[1];
  const float* mask    = (const float*)d_in[2];
  const float* carry   = (const float*)d_in[3];
  const float* Wq      = (const float*)d_in[4];
  const float* Wk      = (const float*)d_in[5];
  const float* Wv      = (const float*)d_in[6];
  const float* Wbeta   = (const float*)d_in[7];
  const float* embed   = (const float*)d_in[8];
  const float* Walpha  = (const float*)d_in[9];
  const float* Wbu     = (const float*)d_in[10];
  const float* Wbv     = (const float*)d_in[11];
  const float* rscale  = (const float*)d_in[12];
  const float* Wout    = (const float*)d_in[13];
  const float* bout    = (const float*)d_in[14];

  const int Bn = 2, Ln = 512, Fn = 512, Hn = 8, Dn = 64;
  const int rows = Bn * Ln;     // 1024
  const int HD   = Hn * Dn;     // 512

  float* ws    = (float*)d_ws;
  float* ae    = ws;                      // rows*HD
  float* qb    = ae   + (size_t)rows * HD;
  float* kb    = qb   + (size_t)rows * HD;
  float* vb    = kb   + (size_t)rows * HD;
  float* bub   = vb   + (size_t)rows * HD;
  float* wb    = bub  + (size_t)rows * HD;   // holds b_v, then w in place
  float* outp  = wb   + (size_t)rows * HD;
  float* rmso  = outp + (size_t)rows * HD;
  float* betab = rmso + (size_t)rows * HD;   // rows*Hn
  float* alphab= betab + (size_t)rows * Hn;

  float* carry_out = (float*)d_out;                       // B*H*D*D
  float* y         = (float*)d_out + (size_t)Bn * Hn * Dn * Dn;

  // 1. gather action embeddings
  k_gather<<<dim3((rows * HD + 255) / 256), dim3(256), 0, stream>>>(action, embed, ae, rows, HD);

  // 2. fp32 WMMA GEMMs (async-LDS double-buffered): q,k,v from x; b_u,b_v from ae
  dim3 ggrid((rows / 128) * (HD / 64));                   // 64 blocks
  k_gemm_wmma<<<ggrid, 256, 0, stream>>>(x,  Wq,  nullptr, qb,  rows, HD, Fn);
  k_gemm_wmma<<<ggrid, 256, 0, stream>>>(x,  Wk,  nullptr, kb,  rows, HD, Fn);
  k_gemm_wmma<<<ggrid, 256, 0, stream>>>(x,  Wv,  nullptr, vb,  rows, HD, Fn);
  k_gemm_wmma<<<ggrid, 256, 0, stream>>>(ae, Wbu, nullptr, bub, rows, HD, Fn);
  k_gemm_wmma<<<ggrid, 256, 0, stream>>>(ae, Wbv, nullptr, wb,  rows, HD, Fn);

  // 3. beta / alpha sigmoid projections (N=8)
  k_proj_sigmoid<<<dim3((rows * Hn + 255) / 256), dim3(256), 0, stream>>>(x,  Wbeta,  betab,  rows, Fn, Hn);
  k_proj_sigmoid<<<dim3((rows * Hn + 255) / 256), dim3(256), 0, stream>>>(ae, Walpha, alphab, rows, HD, Hn);

  // 4. silu + L2-norm q,k; build w = b_v - beta*(b_v.k)*k   (wave per (b,l,h))
  k_prep<<<dim3(rows * Hn * 32 / 256), dim3(256), 0, stream>>>(qb, kb, wb, betab, rows * Hn);

  // 5. sequential scan, 1024 independent row-chains (wave per (b,h,d))
  k_scan<<<dim3(Bn * Hn * Dn * 32 / 256), dim3(256), 0, stream>>>(
      qb, kb, vb, bub, wb, betab, alphab, mask, carry, outp, carry_out, Bn, Ln, Hn, Dn);

  // 6. RMS norm (wave per row)
  k_rms<<<dim3(rows * 32 / 256), dim3(256), 0, stream>>>(outp, rscale, rmso, rows, HD);

  // 7. final projection with bias -> y
  k_gemm_wmma<<<ggrid, 256, 0, stream>>>(rmso, Wout, bout, y, rows, Fn, HD);
}